// VDP_intermediate_Conv_31997506355857
// MI455X (gfx1250) — compile-verified
//
#include <hip/hip_runtime.h>
#include <hip/hip_bf16.h>
#include <math.h>

// Problem constants (from reference)
#define B_  32
#define H_  14
#define W_  14
#define C_  32
#define K_  64
#define KS_ 5
#define OH_ 10
#define OW_ 10
#define P_  100          // OH*OW
#define M_  800          // KS*KS*C
#define N_  196          // H*W
#define MU_OUT_ELEMS (B_ * P_ * K_)   // 204800

typedef __attribute__((ext_vector_type(2))) float v2f;
typedef __attribute__((ext_vector_type(4))) float v4f;
typedef __attribute__((ext_vector_type(8))) float v8f;

// ---------------------------------------------------------------------------
// sp[k] = softplus(w_sigma[k])
// ---------------------------------------------------------------------------
__global__ void sp_kernel(const float* __restrict__ w_sigma, float* __restrict__ sp) {
    int k = threadIdx.x;
    if (k < K_) {
        sp[k] = log1pf(expf(w_sigma[k]));
    }
}

// ---------------------------------------------------------------------------
// diag_vals[b,p,k] = sum_m dsg[b,p,m]*w_mu[m,k]^2 + sp[k] * sum_m dsg[b,p,m]
// dsg pulls ONLY the diagonal of Sigma_in (0.8MB of a 157MB tensor).
// ---------------------------------------------------------------------------
__global__ void diag_vals_kernel(const float* __restrict__ Sigma_in,
                                 const float* __restrict__ w_mu,
                                 const float* __restrict__ sp,
                                 float* __restrict__ diagv) {
    int t = blockIdx.x * blockDim.x + threadIdx.x;
    if (t >= B_ * P_ * K_) return;
    int k = t & (K_ - 1);
    int p = (t / K_) % P_;
    int b = t / (K_ * P_);
    int oy = p / OW_, ox = p % OW_;

    float v1 = 0.f, tr = 0.f;
    #pragma unroll 4
    for (int m = 0; m < M_; ++m) {
        int i   = m / (KS_ * C_);
        int rem = m - i * (KS_ * C_);
        int j   = rem >> 5;          // / C_
        int c   = rem & 31;          // % C_
        int n   = (oy + i) * W_ + (ox + j);
        float s = Sigma_in[(((size_t)b * N_ + n) * N_ + n) * C_ + c];
        float w = w_mu[m * K_ + k];
        v1 += s * w * w;
        tr += s;
    }
    diagv[t] = v1 + sp[k] * tr;
}

// ---------------------------------------------------------------------------
// mu_out[b,p,k] = sum_m mu_p[b,p,m] * w_mu[m,k]   (VALID conv, NHWC/HWIO)
// ---------------------------------------------------------------------------
__global__ void mu_out_kernel(const float* __restrict__ mu_in,
                              const float* __restrict__ w_mu,
                              float* __restrict__ mu_out) {
    int t = blockIdx.x * blockDim.x + threadIdx.x;
    if (t >= B_ * P_ * K_) return;
    int k = t & (K_ - 1);
    int p = (t / K_) % P_;
    int b = t / (K_ * P_);
    int oy = p / OW_, ox = p % OW_;

    const float* base = mu_in + ((size_t)b * H_ * W_ + (size_t)oy * W_ + ox) * C_;
    float acc = 0.f;
    #pragma unroll 4
    for (int m = 0; m < M_; ++m) {
        int i   = m / (KS_ * C_);
        int rem = m - i * (KS_ * C_);
        int j   = rem >> 5;
        int c   = rem & 31;
        acc += base[(i * W_ + j) * C_ + c] * w_mu[m * K_ + k];
    }
    mu_out[t] = acc;
}

// ---------------------------------------------------------------------------
// Main kernel: one wave32 per 16x16 tile of G[b] = mu_p[b] @ mu_p[b]^T using
// V_WMMA_F32_16X16X4_F32 over M=800 (200 WMMAs, 2 accumulators for ILP).
// Fused branch-free epilogue writes Sigma_out exactly once as non-temporal
// b128 stores. DIAG=true handles the tp==tq tiles (only place p==q occurs);
// DIAG=false covers tp!=tq via the skewed mapping tp=(tq+1+y)%T so the
// store-bound output is written exactly once per element.
// ---------------------------------------------------------------------------
template <bool DIAG>
__global__ __launch_bounds__(32)
void sigma_wmma_kernel(const float* __restrict__ mu_in,
                       const float* __restrict__ sp_g,
                       const float* __restrict__ diagv,
                       float* __restrict__ Sigma_out,
                       float* __restrict__ dump) {
    constexpr int T = (P_ + 15) / 16;   // 7 tiles per dim
    __shared__ float s_sp[K_];
    const int lane = threadIdx.x;
    s_sp[lane]      = sp_g[lane];
    s_sp[lane + 32] = sp_g[lane + 32];
    __syncthreads();

    const int tq = blockIdx.x;                                    // 0..6
    const int tp = DIAG ? tq : (int)((tq + 1 + blockIdx.y) % T);  // skewed off-diag
    const int b  = blockIdx.z;                                    // 0..31

    const int rsel = lane & 15;           // row/col within tile
    const int koff = (lane >> 4) << 1;    // 0 or 2: per-lane k offset

    // Clamp padded rows/cols (p,q in [100,112)) to valid memory; padded
    // lanes store to the dump page instead.
    const int p_row = min(tp * 16 + rsel, P_ - 1);
    const int q_col = min(tq * 16 + rsel, P_ - 1);
    // Fold the per-lane k offset into the base pointer so the loop offset
    // stays uniform (scalar) and folds into load immediates after unroll.
    const float* baseA =
        mu_in + ((size_t)b * H_ * W_ + (size_t)(p_row / OW_) * W_ + (p_row % OW_)) * C_ + koff;
    const float* baseB =
        mu_in + ((size_t)b * H_ * W_ + (size_t)(q_col / OW_) * W_ + (q_col % OW_)) * C_ + koff;

    v8f acc0 = {};
    v8f acc1 = {};
    for (int k0 = 0; k0 < M_; k0 += 8) {
        {
            int i   = k0 / (KS_ * C_);              // uniform scalar math
            int rem = k0 - i * (KS_ * C_);
            int off = (i * W_ + (rem >> 5)) * C_ + (rem & 31);
            v2f a  = *(const v2f*)(baseA + off);
            v2f bb = *(const v2f*)(baseB + off);
            acc0 = __builtin_amdgcn_wmma_f32_16x16x4_f32(
                false, a, false, bb, (short)0, acc0, false, false);
        }
        {
            int m0  = k0 + 4;
            int i   = m0 / (KS_ * C_);
            int rem = m0 - i * (KS_ * C_);
            int off = (i * W_ + (rem >> 5)) * C_ + (rem & 31);
            v2f a  = *(const v2f*)(baseA + off);
            v2f bb = *(const v2f*)(baseB + off);
            acc1 = __builtin_amdgcn_wmma_f32_16x16x4_f32(
                false, a, false, bb, (short)0, acc1, false, false);
        }
    }
    v8f acc = acc0 + acc1;

    // ---- Branch-free epilogue ---------------------------------------------
    // C/D layout: VGPR idx holds M=idx (lanes 0-15) / M=idx+8 (lanes 16-31);
    // N = lane & 15.
    const int n_local = lane & 15;
    const int mbase   = (lane >> 4) * 8;
    const int q       = tq * 16 + n_local;
    const bool qvalid = (q < P_);

    // Per-idx output pointers: invalid (p,q) lanes write to the dump page.
    // Offsets are computed unconditionally and selected, so this stays
    // select-based (no exec-mask regions).
    const size_t dump_off = (size_t)(dump - Sigma_out) + (size_t)(lane << 4);
    float* outp[8];
    float  dsel[8];
    #pragma unroll
    for (int idx = 0; idx < 8; ++idx) {
        int    p        = tp * 16 + mbase + idx;
        bool   valid    = qvalid && (p < P_);
        size_t real_off = (((size_t)b * P_ + p) * P_ + (size_t)q) * K_;
        outp[idx]       = Sigma_out + (valid ? real_off : dump_off);
        dsel[idx]       = (DIAG && p == q) ? 1.0f : 0.0f;
    }
    // On a diagonal tile, p==q implies the diag row equals q: one diag
    // float4 load per k-group per lane, blended with an fma selector.
    const float* dvq = diagv + ((size_t)b * P_ + (qvalid ? q : 0)) * K_;

    for (int k = 0; k < K_; k += 4) {
        const v4f spv = *(const v4f*)(s_sp + k);
        v4f dvv = {};
        if (DIAG) dvv = *(const v4f*)(dvq + k);      // uniform-flow load
        const float dva[4] = {dvv.x, dvv.y, dvv.z, dvv.w};
        const float spa[4] = {spv.x, spv.y, spv.z, spv.w};
        #pragma unroll
        for (int idx = 0; idx < 8; ++idx) {
            const float g = acc[idx];
            v4f vv;
            #pragma unroll
            for (int u = 0; u < 4; ++u) {
                float val = spa[u] * g;
                val = fmaf(dsel[idx], dva[u], val);              // diag add (0/1 sel)
                val = __builtin_isfinite(val) ? val : 0.0f;      // nan/inf -> 0
                val = (q == k + u) ? __builtin_fabsf(val) : val; // set_diag abs
                vv[u] = val;
            }
            // Streamed write-once output: non-temporal b128 store.
            __builtin_nontemporal_store(vv, (v4f*)(outp[idx] + k));
        }
    }
}

// ---------------------------------------------------------------------------
extern "C" void kernel_launch(void* const* d_in, const int* in_sizes, int n_in,
                              void* d_out, int out_size, void* d_ws, size_t ws_size,
                              hipStream_t stream) {
    const float* mu_in    = (const float*)d_in[0];
    const float* Sigma_in = (const float*)d_in[1];
    const float* w_mu     = (const float*)d_in[2];
    const float* w_sigma  = (const float*)d_in[3];

    float* out       = (float*)d_out;
    float* mu_out    = out;                 // [32,10,10,64] = 204800 floats
    float* Sigma_out = out + MU_OUT_ELEMS;  // [32,100,100,64] floats

    float* sp    = (float*)d_ws;            // 64 floats
    float* diagv = sp + K_;                 // 204800 floats
    float* dump  = diagv + MU_OUT_ELEMS;    // 512-float dump page for OOB lanes

    sp_kernel<<<1, 64, 0, stream>>>(w_sigma, sp);

    int nt = B_ * P_ * K_;
    diag_vals_kernel<<<(nt + 255) / 256, 256, 0, stream>>>(Sigma_in, w_mu, sp, diagv);
    mu_out_kernel<<<(nt + 255) / 256, 256, 0, stream>>>(mu_in, w_mu, mu_out);

    const int T = (P_ + 15) / 16;  // 7
    // Off-diagonal tiles: tp = (tq + 1 + y) % T, y in [0, T-1) -> each
    // (tp != tq) tile exactly once. Diagonal tiles: DIAG=true, 1D grid.
    dim3 goff(T, T - 1, B_);
    sigma_wmma_kernel<false><<<goff, 32, 0, stream>>>(mu_in, sp, diagv, Sigma_out, dump);
    dim3 gdiag(T, 1, B_);
    sigma_wmma_kernel<true><<<gdiag, 32, 0, stream>>>(mu_in, sp, diagv, Sigma_out, dump);
}